// GCN_41772851920952
// MI455X (gfx1250) — compile-verified
//
#include <hip/hip_runtime.h>
#include <hip/hip_bf16.h>

#define N_NODES 50000
#define IN_F    96
#define HID     64
#define N_CLS   32

typedef __attribute__((ext_vector_type(2))) float v2f;
typedef __attribute__((ext_vector_type(8))) float v8f;

// ---------------------------------------------------------------------------
// Zero a float range (graph-capture-safe init for the scatter accumulators).
// ---------------------------------------------------------------------------
__global__ void zero_f4(float4* __restrict__ p, int n4) {
    int i = blockIdx.x * blockDim.x + threadIdx.x;
    if (i < n4) p[i] = make_float4(0.f, 0.f, 0.f, 0.f);
}

// ---------------------------------------------------------------------------
// Dense GEMM  Y[M,N] = X[M,K] @ W[K,N]  using V_WMMA_F32_16X16X4_F32.
// One wave computes a 16(M) x N strip; K stepped by 4.
// A frag (16x4 f32, 2 VGPR): lanes 0-15 -> M=lane, K = k+0,k+1
//                            lanes 16-31 -> M=lane-16, K = k+2,k+3
// B frag (4x16 f32, 2 VGPR): mirrored (row striped across lanes).
// C (16x16 f32, 8 VGPR): vgpr r, lanes 0-15 -> M=r, N=lane;
//                        lanes 16-31 -> M=r+8, N=lane-16.
// M must be a multiple of 16 (50000 = 3125*16); N,K multiples of 16/4.
// ---------------------------------------------------------------------------
template <int K, int N>
__global__ void gemm_wmma_f32(const float* __restrict__ X,
                              const float* __restrict__ W,
                              float* __restrict__ Y, int M) {
    __shared__ float sW[K * N];
    for (int i = threadIdx.x; i < K * N; i += blockDim.x) sW[i] = W[i];
    __syncthreads();

    const int wave = (int)(blockIdx.x * blockDim.x + threadIdx.x) >> 5;
    const int lane = threadIdx.x & 31;
    const int m0   = wave * 16;
    if (m0 >= M) return;                 // wave-uniform: EXEC stays all-ones

    const int half16 = lane >> 4;        // 0: lanes 0-15, 1: lanes 16-31
    const int l15    = lane & 15;
    const int kh     = half16 * 2;       // K sub-offset for A/B frags
    const int arow   = m0 + l15;

    constexpr int NT = N / 16;
    v8f acc[NT] = {};

    for (int k = 0; k < K; k += 4) {
        v2f a;
        a.x = X[(size_t)arow * K + k + kh + 0];
        a.y = X[(size_t)arow * K + k + kh + 1];
#pragma unroll
        for (int t = 0; t < NT; ++t) {
            v2f b;
            b.x = sW[(k + kh + 0) * N + t * 16 + l15];
            b.y = sW[(k + kh + 1) * N + t * 16 + l15];
            acc[t] = __builtin_amdgcn_wmma_f32_16x16x4_f32(
                false, a, false, b, (short)0, acc[t], false, false);
        }
    }

    const int rbase = m0 + half16 * 8;
#pragma unroll
    for (int t = 0; t < NT; ++t)
#pragma unroll
        for (int r = 0; r < 8; ++r)
            Y[(size_t)(rbase + r) * N + t * 16 + l15] = acc[t][r];
}

// ---------------------------------------------------------------------------
// Edge aggregation: agg[dst[e]][:] += msg[src[e]][:]   (F floats per edge).
// F/4 lanes per edge, float4 gather + 4x GLOBAL_ATOMIC_ADD_F32 scatter.
// ---------------------------------------------------------------------------
template <int F>
__global__ void scatter_add(const float* __restrict__ msg,
                            const int* __restrict__ src,
                            const int* __restrict__ dst,
                            float* __restrict__ agg, int n_edges) {
    constexpr int LPE = F / 4;
    const long long t = (long long)blockIdx.x * blockDim.x + threadIdx.x;
    const int e = (int)(t / LPE);
    const int l = (int)(t % LPE);
    if (e >= n_edges) return;
    const int s = src[e];
    const int d = dst[e];
    const float4 v = *(const float4*)(msg + (size_t)s * F + l * 4);
    float* out = agg + (size_t)d * F + l * 4;
    unsafeAtomicAdd(out + 0, v.x);
    unsafeAtomicAdd(out + 1, v.y);
    unsafeAtomicAdd(out + 2, v.z);
    unsafeAtomicAdd(out + 3, v.w);
}

// ---------------------------------------------------------------------------
// h = relu(agg + b1), 64 features/row, float4 vectorized.
// ---------------------------------------------------------------------------
__global__ void bias_relu64(const float4* __restrict__ agg,
                            const float* __restrict__ b,
                            float4* __restrict__ h, int n4) {
    int i = blockIdx.x * blockDim.x + threadIdx.x;
    if (i >= n4) return;
    float4 v = agg[i];
    const int c = (i & 15) * 4;          // column within 64-wide row
    v.x = fmaxf(v.x + b[c + 0], 0.f);
    v.y = fmaxf(v.y + b[c + 1], 0.f);
    v.z = fmaxf(v.z + b[c + 2], 0.f);
    v.w = fmaxf(v.w + b[c + 3], 0.f);
    h[i] = v;
}

// ---------------------------------------------------------------------------
// out = log_softmax(agg2 + b2) per row; 32 classes == one wave32, lane=class.
// ---------------------------------------------------------------------------
__global__ void logsoftmax32(const float* __restrict__ agg2,
                             const float* __restrict__ b2,
                             float* __restrict__ out, int n_nodes) {
    const int wid  = (int)(blockIdx.x * blockDim.x + threadIdx.x) >> 5;
    const int lane = threadIdx.x & 31;
    if (wid >= n_nodes) return;
    float v = agg2[(size_t)wid * 32 + lane] + b2[lane];
    float m = v;
#pragma unroll
    for (int o = 16; o; o >>= 1) m = fmaxf(m, __shfl_xor(m, o, 32));
    float e = __expf(v - m);
#pragma unroll
    for (int o = 16; o; o >>= 1) e += __shfl_xor(e, o, 32);
    out[(size_t)wid * 32 + lane] = (v - m) - __logf(e);
}

// ---------------------------------------------------------------------------
// Orchestration. Inputs: x, edge_index, W1, b1, W2, b2.
// Workspace layout (floats): t1[50000*64] | t2[50000*32] | agg1[50000*64] | agg2[50000*32]
// ---------------------------------------------------------------------------
extern "C" void kernel_launch(void* const* d_in, const int* in_sizes, int n_in,
                              void* d_out, int out_size, void* d_ws, size_t ws_size,
                              hipStream_t stream) {
    const float* x   = (const float*)d_in[0];
    const int*   ei  = (const int*)d_in[1];
    const float* W1  = (const float*)d_in[2];
    const float* b1  = (const float*)d_in[3];
    const float* W2  = (const float*)d_in[4];
    const float* b2  = (const float*)d_in[5];
    float*       out = (float*)d_out;

    const int n_edges = in_sizes[1] / 2;
    const int* src = ei;
    const int* dst = ei + n_edges;

    float* t1   = (float*)d_ws;                       // 50000*64
    float* t2   = t1 + (size_t)N_NODES * HID;         // 50000*32
    float* agg1 = t2 + (size_t)N_NODES * N_CLS;       // 50000*64
    float* agg2 = agg1 + (size_t)N_NODES * HID;       // 50000*32
    (void)ws_size; (void)n_in; (void)out_size;

    // 0) zero the accumulators (agg1 and agg2 are contiguous: 4.8M floats)
    {
        int n4 = N_NODES * (HID + N_CLS) / 4;
        zero_f4<<<(n4 + 255) / 256, 256, 0, stream>>>((float4*)agg1, n4);
    }
    // 1) t1 = x @ W1           [50000,96]@[96,64]
    {
        int waves = N_NODES / 16;                     // 3125
        gemm_wmma_f32<IN_F, HID><<<(waves + 7) / 8, 256, 0, stream>>>(x, W1, t1, N_NODES);
    }
    // 2) agg1[dst] += t1[src]  (64 floats/edge, 16 lanes/edge)
    {
        long long thr = (long long)n_edges * (HID / 4);
        scatter_add<HID><<<(int)((thr + 255) / 256), 256, 0, stream>>>(t1, src, dst, agg1, n_edges);
    }
    // 3) h = relu(agg1 + b1)   (reuse t1 as h; t1 is dead)
    {
        int n4 = N_NODES * HID / 4;
        bias_relu64<<<(n4 + 255) / 256, 256, 0, stream>>>((const float4*)agg1, b1, (float4*)t1, n4);
    }
    // 4) t2 = h @ W2           [50000,64]@[64,32]
    {
        int waves = N_NODES / 16;
        gemm_wmma_f32<HID, N_CLS><<<(waves + 7) / 8, 256, 0, stream>>>(t1, W2, t2, N_NODES);
    }
    // 5) agg2[dst] += t2[src]  (32 floats/edge, 8 lanes/edge)
    {
        long long thr = (long long)n_edges * (N_CLS / 4);
        scatter_add<N_CLS><<<(int)((thr + 255) / 256), 256, 0, stream>>>(t2, src, dst, agg2, n_edges);
    }
    // 6) out = log_softmax(agg2 + b2), one wave32 per node
    {
        int waves = N_NODES;
        logsoftmax32<<<(waves + 7) / 8, 256, 0, stream>>>(agg2, b2, out, N_NODES);
    }
}